// EffectiveTransformer_84241488544285
// MI455X (gfx1250) — compile-verified
//
#include <hip/hip_runtime.h>

#define B_   64
#define F_   384
#define T_   384
#define H_   8
#define D_   32
#define HD_  256
#define OUT_ 512
#define K_   (F_ * HD_)   // 98304

typedef __attribute__((ext_vector_type(2))) float v2f;
typedef __attribute__((ext_vector_type(4))) float v4f;
typedef __attribute__((ext_vector_type(8))) float v8f;

// ---------------------------------------------------------------------------
// Kernel 1: per-(b,f,h) softmax statistics.
// scores[b,h,f,t] = alpha*y_t + beta, beta constant over t -> cancels.
// alpha = (Wq_h . Wk_h) * x[b,f] + (bq_h . Wk_h)
// s1[b,f,h] = sum_t softmax(alpha*y_t)[t] * y_t   over valid t < to_len[b]
// ---------------------------------------------------------------------------
__global__ __launch_bounds__(256) void attn_s1_kernel(
    const float* __restrict__ from_seq, const float* __restrict__ to_seq,
    const int* __restrict__ to_len,
    const float* __restrict__ Wq, const float* __restrict__ bq,
    const float* __restrict__ Wk,
    float* __restrict__ s1) {
  __shared__ float ylds[T_];
  __shared__ float aC[H_], cC[H_];
  __shared__ float rmax[256], rmin[256];
  const int b = blockIdx.x, tid = threadIdx.x;
  const int tl = to_len[b];

  float mx = -3.4e38f, mn = 3.4e38f;
  for (int t = tid; t < T_; t += 256) {
    float y = to_seq[b * T_ + t];
    ylds[t] = y;
    if (t < tl) { mx = fmaxf(mx, y); mn = fminf(mn, y); }
  }
  rmax[tid] = mx; rmin[tid] = mn;
  if (tid < H_) {
    float a = 0.f, c = 0.f;
    for (int d = 0; d < D_; ++d) {
      float wk = Wk[tid * D_ + d];
      a = fmaf(Wq[tid * D_ + d], wk, a);
      c = fmaf(bq[tid * D_ + d], wk, c);
    }
    aC[tid] = a; cC[tid] = c;
  }
  __syncthreads();
  for (int s = 128; s > 0; s >>= 1) {
    if (tid < s) {
      rmax[tid] = fmaxf(rmax[tid], rmax[tid + s]);
      rmin[tid] = fminf(rmin[tid], rmin[tid + s]);
    }
    __syncthreads();
  }
  const float ymax = rmax[0], ymin = rmin[0];

  for (int r = tid; r < F_ * H_; r += 256) {
    const int f = r >> 3, h = r & 7;
    const float x = from_seq[b * F_ + f];
    const float alpha = fmaf(aC[h], x, cC[h]);
    const float m = (alpha >= 0.f) ? alpha * ymax : alpha * ymin;
    float Z = 0.f, S = 0.f;
    for (int t = 0; t < tl; ++t) {
      float y = ylds[t];
      float e = __expf(fmaf(alpha, y, -m));
      Z += e;
      S = fmaf(e, y, S);
    }
    s1[(size_t)b * (F_ * H_) + r] = S / Z;
  }
}

// ---------------------------------------------------------------------------
// Kernel 2: ctx[b,f,h,d] = gate * (s1*Wv[h,d] + bv[h,d]); one float4/thread.
// ---------------------------------------------------------------------------
__global__ __launch_bounds__(256) void ctx_kernel(
    const float* __restrict__ s1, const float* __restrict__ Wv,
    const float* __restrict__ bv, const int* __restrict__ from_len,
    const unsigned char* __restrict__ q_mask, float* __restrict__ ctx) {
  const int idx = blockIdx.x * 256 + threadIdx.x;   // B*F*H*8 quads
  const int q  = idx & 7;
  const int r  = idx >> 3;
  const int h  = r & 7;
  const int r2 = r >> 3;
  const int f  = r2 % F_;
  const int b  = r2 / F_;
  const float g = (f < from_len[b] && q_mask[b * H_ + h] != 0) ? 1.f : 0.f;
  const float s = s1[(size_t)(b * F_ + f) * H_ + h] * g;
  const int d0 = q * 4;
  v4f o;
#pragma unroll
  for (int j = 0; j < 4; ++j)
    o[j] = fmaf(s, Wv[h * D_ + d0 + j], g * bv[h * D_ + d0 + j]);
  *(v4f*)(ctx + (size_t)idx * 4) = o;
}

// ---------------------------------------------------------------------------
// Kernel 3: split-K WMMA GEMM  partial[kb] += ctx[64,Kchunk] @ Wo[Kchunk,512]
// 512 threads = 16 waves; wave w owns columns [32w, 32w+32).
// A staged in padded LDS (conflict-free), Wo streamed from HBM exactly once.
// ---------------------------------------------------------------------------
#define KC      192
#define NCHUNK  2
#define KSPLIT  256            // K_ / (KC*NCHUNK)
#define APAD    196            // KC + 4 (bank-conflict padding, keeps 16B align)

__global__ __launch_bounds__(512) void gemm_kernel(
    const float* __restrict__ A,        // ctx, [64, K_] row-major
    const float* __restrict__ Wo,       // [K_, 512] row-major
    float* __restrict__ partials) {     // [KSPLIT, 64, 512]
  __shared__ float Alds[64 * APAD];
  const int tid  = threadIdx.x;
  const int kb   = blockIdx.x;
  const int wv   = tid >> 5;
  const int lane = tid & 31;
  const int l16  = lane & 15;
  const int hi   = lane >> 4;            // lanes 16-31 carry K+2/K+3
  const int colbase = wv * 32;

  v8f acc[4][2];
  v8f zero = {};
#pragma unroll
  for (int mt = 0; mt < 4; ++mt)
#pragma unroll
    for (int nt = 0; nt < 2; ++nt) acc[mt][nt] = zero;

  const float* Bcol = Wo + colbase + l16;

  for (int cc = 0; cc < NCHUNK; ++cc) {
    const int kbase = kb * (KC * NCHUNK) + cc * KC;
    if (cc) __syncthreads();
    // stage 64 x KC tile of A into LDS (v4f, coalesced)
    for (int i = tid; i < 64 * (KC / 4); i += 512) {
      const int row = i / (KC / 4);
      const int c4  = i % (KC / 4);
      v4f va = *(const v4f*)(A + (size_t)row * K_ + kbase + c4 * 4);
      *(v4f*)(&Alds[row * APAD + c4 * 4]) = va;
    }
    __syncthreads();

    for (int s = 0; s < KC / 4; ++s) {
      const int kk = s * 4;
      // A fragments: lane<16 -> row mt*16+l16, K = kk..kk+1; lane>=16 -> kk+2..kk+3
      v2f af[4];
#pragma unroll
      for (int mt = 0; mt < 4; ++mt)
        af[mt] = *(const v2f*)(&Alds[(mt * 16 + l16) * APAD + kk + hi * 2]);

      const size_t rb = (size_t)(kbase + kk + hi * 2) * OUT_;
      __builtin_prefetch(Bcol + rb + (size_t)16 * OUT_, 0, 0);
#pragma unroll
      for (int nt = 0; nt < 2; ++nt) {
        v2f bf;
        bf.x = Bcol[rb + nt * 16];
        bf.y = Bcol[rb + OUT_ + nt * 16];
#pragma unroll
        for (int mt = 0; mt < 4; ++mt)
          acc[mt][nt] = __builtin_amdgcn_wmma_f32_16x16x4_f32(
              false, af[mt], false, bf, (short)0, acc[mt][nt], false, false);
      }
    }
  }

  float* P = partials + (size_t)kb * 64 * OUT_;
#pragma unroll
  for (int mt = 0; mt < 4; ++mt)
#pragma unroll
    for (int nt = 0; nt < 2; ++nt) {
      const int col = colbase + nt * 16 + l16;
#pragma unroll
      for (int j = 0; j < 8; ++j) {
        const int row = mt * 16 + j + hi * 8;   // C layout: VGPR j = rows j / j+8
        P[row * OUT_ + col] = acc[mt][nt][j];
      }
    }
}

// ---------------------------------------------------------------------------
// Kernel 4: deterministic split-K reduction + bias + LayerNorm (eps=1e-3).
// ---------------------------------------------------------------------------
__global__ __launch_bounds__(256) void ln_kernel(
    const float* __restrict__ partials, const float* __restrict__ bo,
    const float* __restrict__ gamma, const float* __restrict__ beta,
    float* __restrict__ out) {
  __shared__ float r1[256], r2[256];
  const int b = blockIdx.x, tid = threadIdx.x;
  const int c0 = tid * 2;
  float o0 = 0.f, o1 = 0.f;
  for (int kb = 0; kb < KSPLIT; ++kb) {
    const v2f p = *(const v2f*)(partials + ((size_t)kb * 64 + b) * OUT_ + c0);
    o0 += p.x; o1 += p.y;
  }
  o0 += bo[c0]; o1 += bo[c0 + 1];
  r1[tid] = o0 + o1;
  r2[tid] = fmaf(o0, o0, o1 * o1);
  __syncthreads();
  for (int s = 128; s > 0; s >>= 1) {
    if (tid < s) { r1[tid] += r1[tid + s]; r2[tid] += r2[tid + s]; }
    __syncthreads();
  }
  const float mean = r1[0] * (1.f / OUT_);
  const float var  = r2[0] * (1.f / OUT_) - mean * mean;
  const float inv  = rsqrtf(var + 1e-3f);
  out[b * OUT_ + c0]     = (o0 - mean) * inv * gamma[c0]     + beta[c0];
  out[b * OUT_ + c0 + 1] = (o1 - mean) * inv * gamma[c0 + 1] + beta[c0 + 1];
}

// ---------------------------------------------------------------------------
extern "C" void kernel_launch(void* const* d_in, const int* in_sizes, int n_in,
                              void* d_out, int out_size, void* d_ws, size_t ws_size,
                              hipStream_t stream) {
  const float* from_seq = (const float*)d_in[0];
  const float* to_seq   = (const float*)d_in[1];
  const int*   from_len = (const int*)d_in[2];
  const int*   to_len   = (const int*)d_in[3];
  const unsigned char* q_mask = (const unsigned char*)d_in[4];
  const float* Wq = (const float*)d_in[5];
  const float* bq = (const float*)d_in[6];
  const float* Wk = (const float*)d_in[7];
  // d_in[8] (bk) cancels in softmax and is unused
  const float* Wv = (const float*)d_in[9];
  const float* bv = (const float*)d_in[10];
  const float* Wo = (const float*)d_in[11];
  const float* bo = (const float*)d_in[12];
  const float* gamma = (const float*)d_in[13];
  const float* beta  = (const float*)d_in[14];

  float* out = (float*)d_out;                         // [64,512]
  float* ctx = (float*)d_out + B_ * OUT_;             // [64,384,8,32]
  float* s1  = (float*)d_ws;                          // 196608 floats
  float* partials = (float*)((char*)d_ws + (size_t)B_ * F_ * H_ * 4); // [256,64,512]

  attn_s1_kernel<<<B_, 256, 0, stream>>>(from_seq, to_seq, to_len, Wq, bq, Wk, s1);
  ctx_kernel<<<(B_ * F_ * H_ * 8) / 256, 256, 0, stream>>>(s1, Wv, bv, from_len, q_mask, ctx);
  gemm_kernel<<<KSPLIT, 512, 0, stream>>>(ctx, Wo, partials);
  ln_kernel<<<B_, 256, 0, stream>>>(partials, bo, gamma, beta, out);
}